// GAT_30571577213144
// MI455X (gfx1250) — compile-verified
//
#include <hip/hip_runtime.h>
#include <math.h>

typedef __attribute__((ext_vector_type(2))) float v2f;
typedef __attribute__((ext_vector_type(8))) float v8f;

#define IN_FEATS 512
#define HIDDEN   128
#define KCHUNK   64
#define A_STRIDE 68    // 16 x KCHUNK tile; 68*4B=272B rows -> 16B-aligned B128 writes,
                       // A-frag read bank = (4m+k)%64, distinct for m=0..15 -> conflict-free
#define B_STRIDE 132   // KCHUNK x 128 tile; 132*4B=528B rows -> 16B aligned

// ---------------- CDNA5 async global->LDS copy (ASYNCcnt path) ----------------
__device__ __forceinline__ void async_ld_b128(void* lds_ptr, const void* gptr) {
  // VDST = per-lane LDS byte address (low 32 bits of flat shared pointer),
  // VADDR = 64-bit global address, GV mode (saddr = off).
  uint32_t lds_off = (uint32_t)(uintptr_t)lds_ptr;
  asm volatile("global_load_async_to_lds_b128 %0, %1, off"
               :: "v"(lds_off), "v"(gptr) : "memory");
}
__device__ __forceinline__ void wait_async0() {
  asm volatile("s_wait_asynccnt 0" ::: "memory");
}

// ---------------- init: zero out/denom, set e_max = -inf ----------------
__global__ __launch_bounds__(256) void gat_init(float* __restrict__ out,
                                                float* __restrict__ e_max,
                                                float* __restrict__ denom,
                                                int n_nodes) {
  int i = blockIdx.x * blockDim.x + threadIdx.x;
  int n_out = n_nodes * HIDDEN;
  if (i < n_out) out[i] = 0.0f;
  if (i < n_nodes) { e_max[i] = -INFINITY; denom[i] = 0.0f; }
}

// ---------------- GEMM: x[N,128] = data[N,512] @ W[512,128] via fp32 WMMA ----------------
// Double-buffered: async-copy chunk k+1 into LDS while WMMA consumes chunk k.
__global__ __launch_bounds__(256) void gat_gemm(const float* __restrict__ data,
                                                const float* __restrict__ W,
                                                float* __restrict__ x,
                                                int n_nodes) {
  __shared__ float Alds[2][16 * A_STRIDE];
  __shared__ float Blds[2][KCHUNK * B_STRIDE];

  const int m0   = blockIdx.x * 16;
  const int tid  = threadIdx.x;
  const int wave = tid >> 5;         // 8 waves -> 8 N-tiles of 16
  const int lane = tid & 31;
  const int n0   = wave * 16;

  // fragment lane mapping (ISA 7.12.2, 32-bit 16x4 A / 4x16 B)
  const int mfrag = lane & 15;            // row within A frag / col within B frag
  const int kbase = (lane >> 4) * 2;      // lanes 16-31 hold K+2 / K+3

  // staging coordinates (one B128 for A, eight for B, per thread)
  const int a_i  = tid * 4;
  const int a_m  = a_i >> 6;
  const int a_kk = a_i & 63;
  int a_row = m0 + a_m; if (a_row >= n_nodes) a_row = n_nodes - 1;
  const float* a_src_ptr = data + (size_t)a_row * IN_FEATS + a_kk;

  v8f c = {0.f, 0.f, 0.f, 0.f, 0.f, 0.f, 0.f, 0.f};

  // prologue: stage chunk 0 into buffer 0
  {
    async_ld_b128(&Alds[0][a_m * A_STRIDE + a_kk], a_src_ptr);
    #pragma unroll
    for (int j = 0; j < 8; ++j) {
      int i  = (tid + j * 256) * 4;
      int r  = i >> 7;
      int nn = i & 127;
      async_ld_b128(&Blds[0][r * B_STRIDE + nn], W + (size_t)r * HIDDEN + nn);
    }
  }
  wait_async0();
  __syncthreads();

  int buf = 0;
  for (int kc = 0; kc < IN_FEATS; kc += KCHUNK) {
    const int nkc = kc + KCHUNK;
    if (nkc < IN_FEATS) {
      // async-stage next chunk into the other buffer while we compute
      async_ld_b128(&Alds[buf ^ 1][a_m * A_STRIDE + a_kk], a_src_ptr + nkc);
      #pragma unroll
      for (int j = 0; j < 8; ++j) {
        int i  = (tid + j * 256) * 4;
        int r  = i >> 7;
        int nn = i & 127;
        async_ld_b128(&Blds[buf ^ 1][r * B_STRIDE + nn],
                      W + (size_t)(nkc + r) * HIDDEN + nn);
      }
    }

    const float* Ab = Alds[buf];
    const float* Bb = Blds[buf];
    #pragma unroll
    for (int k = 0; k < KCHUNK; k += 4) {
      v2f a, b;
      a.x = Ab[mfrag * A_STRIDE + k + kbase];
      a.y = Ab[mfrag * A_STRIDE + k + kbase + 1];
      b.x = Bb[(k + kbase) * B_STRIDE + n0 + mfrag];
      b.y = Bb[(k + kbase + 1) * B_STRIDE + n0 + mfrag];
      // 8 args: (neg_a, A, neg_b, B, c_mod, C, reuse_a, reuse_b)
      c = __builtin_amdgcn_wmma_f32_16x16x4_f32(false, a, false, b, (short)0, c,
                                                false, false);
    }

    wait_async0();     // this wave's copies for the next buffer are done
    __syncthreads();   // all waves: compute on buf done + next buffer fully written
    buf ^= 1;
  }

  // C layout: VGPR r -> row r (lanes 0-15) / row r+8 (lanes 16-31), col = lane&15
  const int mhi  = (lane < 16) ? 0 : 8;
  const int ncol = n0 + (lane & 15);
  #pragma unroll
  for (int r = 0; r < 8; ++r) {
    int row = m0 + mhi + r;
    if (row < n_nodes) x[(size_t)row * HIDDEN + ncol] = c[r];
  }
}

// ---------------- per-node attention logits: a = x . att (wave per node) ----------------
__global__ __launch_bounds__(256) void gat_att(const float* __restrict__ x,
                                               const float* __restrict__ att_src,
                                               const float* __restrict__ att_dst,
                                               float* __restrict__ a_src,
                                               float* __restrict__ a_dst,
                                               int n_nodes) {
  int node = (blockIdx.x * blockDim.x + threadIdx.x) >> 5;
  int lane = threadIdx.x & 31;
  if (node >= n_nodes) return;
  float4 xv = *(const float4*)(x + (size_t)node * HIDDEN + lane * 4);
  float4 as = *(const float4*)(att_src + lane * 4);
  float4 ad = *(const float4*)(att_dst + lane * 4);
  float s = xv.x * as.x + xv.y * as.y + xv.z * as.z + xv.w * as.w;
  float d = xv.x * ad.x + xv.y * ad.y + xv.z * ad.z + xv.w * ad.w;
  #pragma unroll
  for (int off = 16; off > 0; off >>= 1) {
    s += __shfl_down(s, off, 32);
    d += __shfl_down(d, off, 32);
  }
  if (lane == 0) { a_src[node] = s; a_dst[node] = d; }
}

// ---------------- deterministic float atomic max (sign-split trick) ----------------
__device__ __forceinline__ void atomic_max_float(float* addr, float val) {
  if (val >= 0.0f) atomicMax((int*)addr, __float_as_int(val));
  else             atomicMin((unsigned int*)addr, __float_as_uint(val));
}

__device__ __forceinline__ void edge_endpoints(const int* adj, int n_edges, int t,
                                               int& s, int& d) {
  if (t < n_edges) { s = adj[t]; d = adj[n_edges + t]; }
  else             { s = t - n_edges; d = s; }   // self loop
}

// ---------------- pass A: segment max of leaky-relu logits ----------------
__global__ __launch_bounds__(256) void gat_edge_max(const int* __restrict__ adj,
                                                    int n_edges, int e_total,
                                                    const float* __restrict__ a_src,
                                                    const float* __restrict__ a_dst,
                                                    float* __restrict__ e_max) {
  int t = blockIdx.x * blockDim.x + threadIdx.x;
  if (t >= e_total) return;
  int s, d; edge_endpoints(adj, n_edges, t, s, d);
  float e = a_src[s] + a_dst[d];
  e = (e > 0.0f) ? e : 0.2f * e;
  atomic_max_float(&e_max[d], e);
}

// ---------------- pass B: segment sum of exp(e - max) ----------------
__global__ __launch_bounds__(256) void gat_edge_den(const int* __restrict__ adj,
                                                    int n_edges, int e_total,
                                                    const float* __restrict__ a_src,
                                                    const float* __restrict__ a_dst,
                                                    const float* __restrict__ e_max,
                                                    float* __restrict__ denom) {
  int t = blockIdx.x * blockDim.x + threadIdx.x;
  if (t >= e_total) return;
  int s, d; edge_endpoints(adj, n_edges, t, s, d);
  float e = a_src[s] + a_dst[d];
  e = (e > 0.0f) ? e : 0.2f * e;
  unsafeAtomicAdd(&denom[d], __expf(e - e_max[d]));
}

// ---------------- pass C: weighted scatter, wave per edge, float4 per lane ----------------
__global__ __launch_bounds__(256) void gat_edge_agg(const int* __restrict__ adj,
                                                    int n_edges, int e_total,
                                                    const float* __restrict__ a_src,
                                                    const float* __restrict__ a_dst,
                                                    const float* __restrict__ e_max,
                                                    const float* __restrict__ denom,
                                                    const float* __restrict__ x,
                                                    float* __restrict__ out) {
  int t    = (blockIdx.x * blockDim.x + threadIdx.x) >> 5;
  int lane = threadIdx.x & 31;
  if (t >= e_total) return;
  int s, d; edge_endpoints(adj, n_edges, t, s, d);
  float e = a_src[s] + a_dst[d];
  e = (e > 0.0f) ? e : 0.2f * e;
  float alpha = __expf(e - e_max[d]) / (denom[d] + 1e-16f);
  float4 xv = *(const float4*)(x + (size_t)s * HIDDEN + lane * 4);  // L2-resident gather
  float* o = out + (size_t)d * HIDDEN + lane * 4;
  unsafeAtomicAdd(o + 0, alpha * xv.x);
  unsafeAtomicAdd(o + 1, alpha * xv.y);
  unsafeAtomicAdd(o + 2, alpha * xv.z);
  unsafeAtomicAdd(o + 3, alpha * xv.w);
}

// ---------------- finalize: bias + leaky_relu(0.25) in place ----------------
__global__ __launch_bounds__(256) void gat_final(float* __restrict__ out,
                                                 const float* __restrict__ bias,
                                                 int n_out) {
  int i = blockIdx.x * blockDim.x + threadIdx.x;
  if (i >= n_out) return;
  float v = out[i] + bias[i & (HIDDEN - 1)];
  out[i] = (v > 0.0f) ? v : 0.25f * v;
}

extern "C" void kernel_launch(void* const* d_in, const int* in_sizes, int n_in,
                              void* d_out, int out_size, void* d_ws, size_t ws_size,
                              hipStream_t stream) {
  const float* data    = (const float*)d_in[0];
  const int*   adj     = (const int*)d_in[1];
  const float* W       = (const float*)d_in[2];
  const float* att_src = (const float*)d_in[3];
  const float* att_dst = (const float*)d_in[4];
  const float* bias    = (const float*)d_in[5];
  float*       out     = (float*)d_out;

  const int n_nodes = in_sizes[0] / IN_FEATS;  // 100000
  const int n_edges = in_sizes[1] / 2;         // 3200000
  const int e_total = n_edges + n_nodes;       // incl. self loops
  const int n_out   = n_nodes * HIDDEN;

  // workspace carve-up (16B aligned)
  char* ws = (char*)d_ws;
  size_t off = 0;
  float* x      = (float*)(ws + off); off += (size_t)n_nodes * HIDDEN * sizeof(float);
  float* a_src  = (float*)(ws + off); off += ((size_t)n_nodes * sizeof(float) + 15) & ~15ull;
  float* a_dst  = (float*)(ws + off); off += ((size_t)n_nodes * sizeof(float) + 15) & ~15ull;
  float* e_max  = (float*)(ws + off); off += ((size_t)n_nodes * sizeof(float) + 15) & ~15ull;
  float* denom  = (float*)(ws + off); off += ((size_t)n_nodes * sizeof(float) + 15) & ~15ull;
  (void)ws_size; (void)n_in; (void)out_size;

  const int B = 256;
  gat_init<<<(n_out + B - 1) / B, B, 0, stream>>>(out, e_max, denom, n_nodes);
  gat_gemm<<<(n_nodes + 15) / 16, B, 0, stream>>>(data, W, x, n_nodes);
  gat_att<<<((n_nodes * 32) + B - 1) / B, B, 0, stream>>>(x, att_src, att_dst,
                                                          a_src, a_dst, n_nodes);
  gat_edge_max<<<(e_total + B - 1) / B, B, 0, stream>>>(adj, n_edges, e_total,
                                                        a_src, a_dst, e_max);
  gat_edge_den<<<(e_total + B - 1) / B, B, 0, stream>>>(adj, n_edges, e_total,
                                                        a_src, a_dst, e_max, denom);
  {
    long long threads = (long long)e_total * 32;
    int blocks = (int)((threads + B - 1) / B);
    gat_edge_agg<<<blocks, B, 0, stream>>>(adj, n_edges, e_total, a_src, a_dst,
                                           e_max, denom, x, out);
  }
  gat_final<<<(n_out + B - 1) / B, B, 0, stream>>>(out, bias, n_out);
}